// DecoderRNN_3083786518737
// MI455X (gfx1250) — compile-verified
//
#include <hip/hip_runtime.h>
#include <math.h>

// ---------------------------------------------------------------------------
// Problem constants (fixed by the reference harness)
// ---------------------------------------------------------------------------
#define B_   64      // batch
#define E_   512     // embed dim
#define D_   512     // decoder hidden dim
#define V_   10000   // vocab
#define S_   32      // caption length
#define T_   31      // decode steps (caption_lengths[0] forced to S -> max dl = 31)
#define G4   2048    // 4*D gates
#define NT_CLS ((T_ * B_ / 16) * (V_ / 16))   // 124*625 = 77500 classifier C-tiles

typedef float v2f __attribute__((ext_vector_type(2)));
typedef float v8f __attribute__((ext_vector_type(8)));

// V_WMMA_F32_16X16X4_F32: D(16x16 f32) = A(16x4 f32) x B(4x16 f32) + C
// 8-arg pattern per probe: (neg_a, A, neg_b, B, c_mod, C, reuse_a, reuse_b)
__device__ __forceinline__ v8f wmma4(v2f a, v2f b, v8f c) {
  return __builtin_amdgcn_wmma_f32_16x16x4_f32(false, a, false, b, (short)0, c,
                                               false, false);
}

__device__ __forceinline__ float sigf(float x) { return 1.0f / (1.0f + expf(-x)); }

// ---------------------------------------------------------------------------
// 1) Gather teacher-forcing embeddings: xs[t][b][k] = embedding[cap[b][t]][k]
// ---------------------------------------------------------------------------
__global__ __launch_bounds__(256) void gather_emb_kernel(
    const float* __restrict__ emb, const int* __restrict__ caps,
    float* __restrict__ xs) {
  int idx = blockIdx.x * 256 + threadIdx.x;            // over T*B*E
  if (idx >= T_ * B_ * E_) return;
  int k  = idx & (E_ - 1);
  int tb = idx >> 9;                                    // /E_
  int b  = tb & (B_ - 1);
  int t  = tb >> 6;                                     // /B_
  int tok = caps[b * S_ + t];
  xs[idx] = emb[(size_t)tok * E_ + k];
}

// ---------------------------------------------------------------------------
// 2) h0 = c0 = encoder_out
// ---------------------------------------------------------------------------
__global__ __launch_bounds__(256) void init_hc_kernel(
    const float* __restrict__ enc, float* __restrict__ hbuf,
    float* __restrict__ cbuf) {
  int i = blockIdx.x * 256 + threadIdx.x;
  if (i < B_ * D_) { hbuf[i] = enc[i]; cbuf[i] = enc[i]; }
}

// ---------------------------------------------------------------------------
// 3) Per-row dot products (time-invariant): wh0, wc0, EW[b][g]=enc[b]·w_wih[g]
//    One wave per batch row, shfl reduction.
// ---------------------------------------------------------------------------
__global__ __launch_bounds__(256) void init_vec_kernel(
    const float* __restrict__ enc, const float* __restrict__ awih,
    const float* __restrict__ ihw, const float* __restrict__ ihb,
    const float* __restrict__ icw, const float* __restrict__ icb,
    float* __restrict__ EW, float* __restrict__ whwc) {
  int gwave = (blockIdx.x * 256 + threadIdx.x) >> 5;    // 0..63 = batch row
  int lane  = threadIdx.x & 31;
  if (gwave >= B_) return;
  const float* e = enc + (size_t)gwave * E_;
  float s0 = 0.f, s1 = 0.f, s2 = 0.f, s3 = 0.f, s4 = 0.f, s5 = 0.f;
  for (int k = lane; k < E_; k += 32) {
    float ev = e[k];
    s0 += ev * ihw[k];
    s1 += ev * icw[k];
    s2 += ev * awih[0 * E_ + k];
    s3 += ev * awih[1 * E_ + k];
    s4 += ev * awih[2 * E_ + k];
    s5 += ev * awih[3 * E_ + k];
  }
  #pragma unroll
  for (int off = 16; off > 0; off >>= 1) {
    s0 += __shfl_xor(s0, off, 32);
    s1 += __shfl_xor(s1, off, 32);
    s2 += __shfl_xor(s2, off, 32);
    s3 += __shfl_xor(s3, off, 32);
    s4 += __shfl_xor(s4, off, 32);
    s5 += __shfl_xor(s5, off, 32);
  }
  if (lane == 0) {
    whwc[gwave]      = s0 + ihb[0];
    whwc[B_ + gwave] = s1 + icb[0];
    EW[gwave * 4 + 0] = s2;
    EW[gwave * 4 + 1] = s3;
    EW[gwave * 4 + 2] = s4;
    EW[gwave * 4 + 3] = s5;
  }
}

// ---------------------------------------------------------------------------
// 4) Time-invariant gate base: EG[b][j] = enc[b]·dec_wih[j,512:] + bih[j]+bhh[j]
//    GEMM M=64,N=2048,K=512 via WMMA f32 16x16x4.
//    Wave w -> M-tile w; block -> 16 columns of each of the 4 gate groups.
// ---------------------------------------------------------------------------
__global__ __launch_bounds__(128) void eg_kernel(
    const float* __restrict__ enc, const float* __restrict__ wih,
    const float* __restrict__ bih, const float* __restrict__ bhh,
    float* __restrict__ EG) {
  const int lane = threadIdx.x & 31;
  const int wave = threadIdx.x >> 5;
  const int half = lane >> 4, l16 = lane & 15;
  const int m0 = wave * 16;
  const int n0 = blockIdx.x * 16;
  v8f acc[4];
  #pragma unroll
  for (int g = 0; g < 4; ++g) {
    int j = g * D_ + n0 + l16;
    float bb = bih[j] + bhh[j];
    #pragma unroll
    for (int r = 0; r < 8; ++r) acc[g][r] = bb;
  }
  const int arow = m0 + l16;
  for (int k = 0; k < E_; k += 4) {
    v2f a;
    a.x = enc[arow * E_ + k + 2 * half];
    a.y = enc[arow * E_ + k + 2 * half + 1];
    #pragma unroll
    for (int g = 0; g < 4; ++g) {
      const float* wr = wih + (size_t)(g * D_ + n0 + l16) * (2 * E_) + E_;
      v2f b;
      b.x = wr[k + 2 * half];
      b.y = wr[k + 2 * half + 1];
      acc[g] = wmma4(a, b, acc[g]);
    }
  }
  #pragma unroll
  for (int g = 0; g < 4; ++g)
    #pragma unroll
    for (int r = 0; r < 8; ++r) {
      int m = m0 + r + 8 * half;
      EG[(size_t)m * G4 + g * D_ + n0 + l16] = acc[g][r];
    }
}

// ---------------------------------------------------------------------------
// 5) Attention LSTM (hidden dim 1): entire 31-step recurrence, 64 lanes.
// ---------------------------------------------------------------------------
__global__ __launch_bounds__(64) void att_kernel(
    const float* __restrict__ EW, const float* __restrict__ whwc,
    const float* __restrict__ awhh, const float* __restrict__ abih,
    const float* __restrict__ abhh, const int* __restrict__ clen,
    float* __restrict__ wseq) {
  int b = threadIdx.x;
  if (b >= B_) return;
  float wh = whwc[b], wc = whwc[B_ + b];
  float e0 = EW[b * 4 + 0], e1 = EW[b * 4 + 1], e2 = EW[b * 4 + 2], e3 = EW[b * 4 + 3];
  float u0 = awhh[0], u1 = awhh[1], u2 = awhh[2], u3 = awhh[3];
  float b0 = abih[0] + abhh[0], b1 = abih[1] + abhh[1];
  float b2 = abih[2] + abhh[2], b3 = abih[3] + abhh[3];
  int dl = clen[b] - 1;
  for (int t = 0; t < T_; ++t) {
    float gi = e0 + wh * u0 + b0;
    float gf = e1 + wh * u1 + b1;
    float gg = e2 + wh * u2 + b2;
    float go = e3 + wh * u3 + b3;
    float cn = sigf(gf) * wc + sigf(gi) * tanhf(gg);
    float hn = sigf(go) * tanhf(cn);
    float mask = (dl > t) ? 1.0f : 0.0f;
    wseq[t * B_ + b] = sigf(hn) * mask;
    if (dl > t) { wh = hn; wc = cn; }
  }
}

// ---------------------------------------------------------------------------
// 6) One LSTM step: gates = EG + xs_t@Wih[:, :512].T + h@Whh.T  (WMMA f32)
//    Then fused epilogue: c2/h2, ragged-batch mask, double-buffered h.
//    Grid: 32 blocks x 4 waves; wave = M-tile, block = 16 gate columns.
// ---------------------------------------------------------------------------
__global__ __launch_bounds__(128) void lstm_step_kernel(
    const float* __restrict__ xs_t, const float* __restrict__ h_in,
    float* __restrict__ h_out, float* __restrict__ cst,
    const float* __restrict__ EG, const float* __restrict__ wih,
    const float* __restrict__ whh, const int* __restrict__ clen,
    float* __restrict__ hall_t, int t) {
  const int lane = threadIdx.x & 31;
  const int wave = threadIdx.x >> 5;
  const int half = lane >> 4, l16 = lane & 15;
  const int m0 = wave * 16;
  const int n0 = blockIdx.x * 16;
  v8f acc[4];
  #pragma unroll
  for (int g = 0; g < 4; ++g)
    #pragma unroll
    for (int r = 0; r < 8; ++r)
      acc[g][r] = EG[(size_t)(m0 + r + 8 * half) * G4 + g * D_ + n0 + l16];

  const int arow = m0 + l16;
  // K in [0,512): A = token embedding, B^T = dec_wih[:, :512] (ld 1024)
  for (int k = 0; k < E_; k += 4) {
    v2f a;
    a.x = xs_t[arow * E_ + k + 2 * half];
    a.y = xs_t[arow * E_ + k + 2 * half + 1];
    #pragma unroll
    for (int g = 0; g < 4; ++g) {
      const float* wr = wih + (size_t)(g * D_ + n0 + l16) * (2 * E_);
      v2f b;
      b.x = wr[k + 2 * half];
      b.y = wr[k + 2 * half + 1];
      acc[g] = wmma4(a, b, acc[g]);
    }
  }
  // K in [512,1024): A = previous h, B^T = dec_whh (ld 512)
  for (int k = 0; k < D_; k += 4) {
    v2f a;
    a.x = h_in[arow * D_ + k + 2 * half];
    a.y = h_in[arow * D_ + k + 2 * half + 1];
    #pragma unroll
    for (int g = 0; g < 4; ++g) {
      const float* wr = whh + (size_t)(g * D_ + n0 + l16) * D_;
      v2f b;
      b.x = wr[k + 2 * half];
      b.y = wr[k + 2 * half + 1];
      acc[g] = wmma4(a, b, acc[g]);
    }
  }
  // fused LSTM epilogue; each (m,n) owned by exactly one lane/VGPR -> c in place
  #pragma unroll
  for (int r = 0; r < 8; ++r) {
    int m = m0 + r + 8 * half;
    int n = n0 + l16;
    float gi = acc[0][r], gf = acc[1][r], gg = acc[2][r], go = acc[3][r];
    float cold = cst[m * D_ + n];
    float c2 = sigf(gf) * cold + sigf(gi) * tanhf(gg);
    float h2 = sigf(go) * tanhf(c2);
    bool keep = (clen[m] - 1) > t;
    h_out[m * D_ + n] = keep ? h2 : h_in[m * D_ + n];
    cst[m * D_ + n]   = keep ? c2 : cold;
    hall_t[m * D_ + n] = h2;   // preds mask applied in classifier
  }
}

// ---------------------------------------------------------------------------
// 7) Batched classifier: preds = Hall[1984,512] @ cls_w.T + cls_b, masked,
//    scattered to [B,T,V]. One wave per 16x16 C-tile, K-loop of 128 WMMAs.
// ---------------------------------------------------------------------------
__global__ __launch_bounds__(128) void cls_kernel(
    const float* __restrict__ hall, const float* __restrict__ clsw,
    const float* __restrict__ clsb, const int* __restrict__ clen,
    float* __restrict__ preds) {
  const int lane = threadIdx.x & 31;
  const int wave = threadIdx.x >> 5;
  const int half = lane >> 4, l16 = lane & 15;
  int tile = blockIdx.x * 4 + wave;
  if (tile >= NT_CLS) return;
  const int NTN = V_ / 16;                 // 625
  int mt = tile / NTN, nt = tile % NTN;
  int m0 = mt * 16, n0 = nt * 16;
  float bn = clsb[n0 + l16];
  v8f acc;
  #pragma unroll
  for (int r = 0; r < 8; ++r) acc[r] = bn;  // bias folded into C init
  for (int k = 0; k < D_; k += 4) {
    v2f a;
    a.x = hall[(size_t)(m0 + l16) * D_ + k + 2 * half];
    a.y = hall[(size_t)(m0 + l16) * D_ + k + 2 * half + 1];
    v2f b;
    const float* wr = clsw + (size_t)(n0 + l16) * D_;
    b.x = wr[k + 2 * half];
    b.y = wr[k + 2 * half + 1];
    acc = wmma4(a, b, acc);
  }
  #pragma unroll
  for (int r = 0; r < 8; ++r) {
    int m = m0 + r + 8 * half;             // m = t*64 + b
    int n = n0 + l16;
    int t = m >> 6;
    int bb = m & 63;
    float v = ((clen[bb] - 1) > t) ? acc[r] : 0.0f;
    preds[((size_t)bb * T_ + t) * V_ + n] = v;
  }
}

// ---------------------------------------------------------------------------
// 8) weights[b][t][v] = wseq[t][b] broadcast; also emit dl as float.
// ---------------------------------------------------------------------------
__global__ __launch_bounds__(256) void bcast_w_kernel(
    const float* __restrict__ wseq, const int* __restrict__ clen,
    float* __restrict__ wout, float* __restrict__ dlo) {
  int idx = blockIdx.x * 256 + threadIdx.x;
  if (idx < B_ && blockIdx.x == 0) dlo[idx] = (float)(clen[idx] - 1);
  if (idx >= B_ * T_ * V_) return;
  int v  = idx % V_;
  int bt = idx / V_;
  int t  = bt % T_;
  int b  = bt / T_;
  (void)v;
  wout[idx] = wseq[t * B_ + b];
}

// ---------------------------------------------------------------------------
// launch
// ---------------------------------------------------------------------------
extern "C" void kernel_launch(void* const* d_in, const int* in_sizes, int n_in,
                              void* d_out, int out_size, void* d_ws,
                              size_t ws_size, hipStream_t stream) {
  (void)in_sizes; (void)n_in; (void)out_size; (void)ws_size;
  const float* enc  = (const float*)d_in[0];
  const int*   caps = (const int*)d_in[1];
  const int*   clen = (const int*)d_in[2];
  const float* emb  = (const float*)d_in[3];
  const float* wih  = (const float*)d_in[4];
  const float* whh  = (const float*)d_in[5];
  const float* bih  = (const float*)d_in[6];
  const float* bhh  = (const float*)d_in[7];
  const float* awih = (const float*)d_in[8];
  const float* awhh = (const float*)d_in[9];
  const float* abih = (const float*)d_in[10];
  const float* abhh = (const float*)d_in[11];
  const float* ihw  = (const float*)d_in[12];
  const float* ihb  = (const float*)d_in[13];
  const float* icw  = (const float*)d_in[14];
  const float* icb  = (const float*)d_in[15];
  const float* clsw = (const float*)d_in[16];
  const float* clsb = (const float*)d_in[17];

  float* preds = (float*)d_out;                       // [B,T,V]
  float* dlo   = preds + (size_t)B_ * T_ * V_;        // [B]
  float* wout  = dlo + B_;                            // [B,T,V]

  // workspace layout (~8.7 MB of floats)
  float* ws   = (float*)d_ws;
  float* xs   = ws;                                   // T*B*E
  float* EG   = xs + (size_t)T_ * B_ * E_;            // B*G4
  float* hbuf = EG + (size_t)B_ * G4;                 // 2*B*D (double buffer)
  float* cbuf = hbuf + 2 * (size_t)B_ * D_;           // B*D
  float* hall = cbuf + (size_t)B_ * D_;               // T*B*D
  float* wseq = hall + (size_t)T_ * B_ * D_;          // T*B
  float* EW   = wseq + T_ * B_;                       // 4*B
  float* whwc = EW + 4 * B_;                          // 2*B

  gather_emb_kernel<<<(T_ * B_ * E_ + 255) / 256, 256, 0, stream>>>(emb, caps, xs);
  init_hc_kernel<<<(B_ * D_ + 255) / 256, 256, 0, stream>>>(enc, hbuf, cbuf);
  init_vec_kernel<<<8, 256, 0, stream>>>(enc, awih, ihw, ihb, icw, icb, EW, whwc);
  eg_kernel<<<D_ / 16, 128, 0, stream>>>(enc, wih, bih, bhh, EG);
  att_kernel<<<1, 64, 0, stream>>>(EW, whwc, awhh, abih, abhh, clen, wseq);

  for (int t = 0; t < T_; ++t) {
    const float* hin = hbuf + (size_t)(t & 1) * (B_ * D_);
    float* hout      = hbuf + (size_t)((t + 1) & 1) * (B_ * D_);
    lstm_step_kernel<<<D_ / 16, 128, 0, stream>>>(
        xs + (size_t)t * B_ * E_, hin, hout, cbuf, EG, wih, whh, clen,
        hall + (size_t)t * B_ * D_, t);
  }

  cls_kernel<<<NT_CLS / 4, 128, 0, stream>>>(hall, clsw, clsb, clen, preds);
  bcast_w_kernel<<<(B_ * T_ * V_ + 255) / 256, 256, 0, stream>>>(wseq, clen, wout, dlo);
}